// CombinedLoss1_87359634801124
// MI455X (gfx1250) — compile-verified
//
#include <hip/hip_runtime.h>
#include <math.h>

// ---------------- geometry ----------------
#define DD 384
#define HH 128
#define WW 64
#define PADR 5
#define DP (DD + 2*PADR)          // 394
#define HP (HH + 2*PADR)          // 138
#define NLINES (DP * HP)          // 54372
#define LTILES ((NLINES + 15)/16) // 3399
#define VOL ((size_t)DD*HH*WW)    // 3,145,728
#define B1_ELEMS ((size_t)DP*HP*WW)   // 3,479,808
#define B2_ELEMS ((size_t)DP*HH*WW)   // 3,227,648
#define RED_OFF (2*B1_ELEMS + 5*B2_ELEMS)
// reduction slot layout (floats, relative to red base)
#define R_DIFFD 0
#define R_M2D   384
#define R_DIFFH 768
#define R_M2H   896
#define R_DIFFW 1024
#define R_M2W   1088
#define R_SCAL  1152   // [0]=inter [1]=sum_pred2 [2]=sum_true2 [3]=sq [4]=ssim_sum
#define R_GTAB  1160   // 28x16 banded Gaussian matrix G[k][p] = g[k-p]
#define GTAB_N  (28*16)

typedef __attribute__((ext_vector_type(2))) float v2f;
typedef __attribute__((ext_vector_type(8))) float v8f;

struct G11 { float g[11]; };

__device__ __forceinline__ int refl(int i, int n) {
  i = (i < 0) ? -i : i;
  return (i >= n) ? (2*n - 2 - i) : i;
}

// compile-time field selection: 0=x, 1=y, 2=x*x, 3=y*y, 4=x*y
template <int MODE>
__device__ __forceinline__ float fieldval(float xv, float yv) {
  if (MODE == 0) return xv;
  if (MODE == 1) return yv;
  if (MODE == 2) return xv * xv;
  if (MODE == 3) return yv * yv;
  return xv * yv;
}

// Build the 28x16 banded filter matrix once per launch (tiny, L2-resident).
// G[k][p] = g[k-p] for 0 <= k-p <= 10, else 0.  2*sigma^2 = 4.5.
__global__ void fill_gtab(float* __restrict__ gtab, float invn) {
  const int i = blockIdx.x * blockDim.x + threadIdx.x;
  if (i < GTAB_N) {
    const int k = i >> 4, p = i & 15;
    const int d = k - p;
    const float t = (float)(d - 5);
    const float v = __expf(-t * t * (1.0f/4.5f)) * invn;
    gtab[i] = (d >= 0 && d <= 10) ? v : 0.0f;
  }
}

// Preload the per-lane A (filter) operands for all 7 K-chunks.
// A VGPR layout (32-bit 16x4): v0: lanes0-15 K=0 / lanes16-31 K=2; v1: K=1 / K=3
__device__ __forceinline__ void load_avals(const float* __restrict__ gtab,
                                           int laneLo, int laneHi, v2f (&av)[7]) {
#pragma unroll
  for (int c = 0; c < 7; ++c) {
    const int k0 = (c << 2) + (laneHi << 1);
    av[c].x = gtab[(k0    ) * 16 + laneLo];   // lanes 0-15: consecutive -> coalesced
    av[c].y = gtab[(k0 + 1) * 16 + laneLo];
  }
}

// ---------------- pass 1: reflect-padded blur along W (WMMA) ----------------
// Out[p, l] tile (16 output-w positions x 16 lines), D = sum_c A_c(16x4 of G^T) * B_c(4x16 input)
// B VGPR layout (32-bit 4x16): v0: lanes0-15 row0 / lanes16-31 row2; v1: row1 / row3
// D layout: lane N=laneLo, VGPR v -> M = v + 8*laneHi
// Dual-field version: two accumulators share the A (filter) operand and the x/y loads.
template <int MA, int MB>
__global__ __launch_bounds__(32) void blur_w_wmma2(const float* __restrict__ xs,
                                                   const float* __restrict__ ys,
                                                   float* __restrict__ outA,
                                                   float* __restrict__ outB,
                                                   const float* __restrict__ gtab) {
  const int lane = threadIdx.x;
  const int laneLo = lane & 15, laneHi = lane >> 4;
  const int w0 = blockIdx.x << 4;        // output-w tile base (0,16,32,48)
  const int lbase = blockIdx.y << 4;     // line tile base
  int line = lbase + laneLo; if (line >= NLINES) line = NLINES - 1;
  const int dp = line / HP, hp = line - dp * HP;
  const int sd = refl(dp - PADR, DD), sh = refl(hp - PADR, HH);
  const float* xrow = xs + ((size_t)sd * HH + sh) * WW;
  const float* yrow = ys + ((size_t)sd * HH + sh) * WW;
  __builtin_prefetch(xrow, 0, 0);
  __builtin_prefetch(yrow, 0, 0);
  v2f av[7];
  load_avals(gtab, laneLo, laneHi, av);
  v8f accA = {0,0,0,0,0,0,0,0};
  v8f accB = {0,0,0,0,0,0,0,0};
#pragma unroll
  for (int c = 0; c < 7; ++c) {
    const int k0 = (c << 2) + (laneHi << 1);
    int we0 = w0 + k0;     if (we0 > 73) we0 = 73;   // rows 26,27 of G are zero anyway
    int we1 = w0 + k0 + 1; if (we1 > 73) we1 = 73;
    const int sw0 = refl(we0 - PADR, WW);
    const int sw1 = refl(we1 - PADR, WW);
    const float x0 = xrow[sw0], y0v = yrow[sw0];
    const float x1 = xrow[sw1], y1v = yrow[sw1];
    v2f bA; bA.x = fieldval<MA>(x0, y0v); bA.y = fieldval<MA>(x1, y1v);
    v2f bB; bB.x = fieldval<MB>(x0, y0v); bB.y = fieldval<MB>(x1, y1v);
    accA = __builtin_amdgcn_wmma_f32_16x16x4_f32(false, av[c], false, bA, (short)0, accA,
                                                 false, false);
    accB = __builtin_amdgcn_wmma_f32_16x16x4_f32(false, av[c], false, bB, (short)0, accB,
                                                 false, false);
  }
  const int lout = lbase + laneLo;
  if (lout < NLINES) {
#pragma unroll
    for (int v = 0; v < 8; ++v) {
      const int wo = w0 + v + (laneHi << 3);
      outA[(size_t)lout * WW + wo] = accA[v];
      outB[(size_t)lout * WW + wo] = accB[v];
    }
  }
}

// Single-field version (used for the x*y cross term).
template <int MODE>
__global__ __launch_bounds__(32) void blur_w_wmma1(const float* __restrict__ xs,
                                                   const float* __restrict__ ys,
                                                   float* __restrict__ out,
                                                   const float* __restrict__ gtab) {
  const int lane = threadIdx.x;
  const int laneLo = lane & 15, laneHi = lane >> 4;
  const int w0 = blockIdx.x << 4;
  const int lbase = blockIdx.y << 4;
  int line = lbase + laneLo; if (line >= NLINES) line = NLINES - 1;
  const int dp = line / HP, hp = line - dp * HP;
  const int sd = refl(dp - PADR, DD), sh = refl(hp - PADR, HH);
  const float* xrow = xs + ((size_t)sd * HH + sh) * WW;
  const float* yrow = ys + ((size_t)sd * HH + sh) * WW;
  __builtin_prefetch(xrow, 0, 0);
  __builtin_prefetch(yrow, 0, 0);
  v2f av[7];
  load_avals(gtab, laneLo, laneHi, av);
  v8f acc = {0,0,0,0,0,0,0,0};
#pragma unroll
  for (int c = 0; c < 7; ++c) {
    const int k0 = (c << 2) + (laneHi << 1);
    int we0 = w0 + k0;     if (we0 > 73) we0 = 73;
    int we1 = w0 + k0 + 1; if (we1 > 73) we1 = 73;
    const int sw0 = refl(we0 - PADR, WW);
    const int sw1 = refl(we1 - PADR, WW);
    v2f b;
    b.x = fieldval<MODE>(xrow[sw0], yrow[sw0]);
    b.y = fieldval<MODE>(xrow[sw1], yrow[sw1]);
    acc = __builtin_amdgcn_wmma_f32_16x16x4_f32(false, av[c], false, b, (short)0, acc,
                                                false, false);
  }
  const int lout = lbase + laneLo;
  if (lout < NLINES) {
#pragma unroll
    for (int v = 0; v < 8; ++v) {
      const int wo = w0 + v + (laneHi << 3);
      out[(size_t)lout * WW + wo] = acc[v];
    }
  }
}

// ---------------- pass 2: blur along H (WMMA), fully coalesced ----------------
// input (DP, HP, WW) -> output (DP, HH, WW)
__global__ __launch_bounds__(32) void blur_h_wmma(const float* __restrict__ in,
                                                  float* __restrict__ out,
                                                  const float* __restrict__ gtab) {
  const int lane = threadIdx.x;
  const int laneLo = lane & 15, laneHi = lane >> 4;
  const int wt = blockIdx.x & 3, ht = blockIdx.x >> 2;
  const int dp = blockIdx.y;
  const int w0 = wt << 4, h0 = ht << 4;
  const float* base = in + (size_t)dp * HP * WW;
  __builtin_prefetch(base + (size_t)h0 * WW + w0, 0, 0);
  v2f av[7];
  load_avals(gtab, laneLo, laneHi, av);
  v8f acc = {0,0,0,0,0,0,0,0};
#pragma unroll
  for (int c = 0; c < 7; ++c) {
    const int k0 = (c << 2) + (laneHi << 1);
    int h1 = h0 + k0;     if (h1 > HP - 1) h1 = HP - 1;
    int h2 = h0 + k0 + 1; if (h2 > HP - 1) h2 = HP - 1;
    v2f b;
    b.x = base[(size_t)h1 * WW + w0 + laneLo];   // 16 consecutive floats/halfwave
    b.y = base[(size_t)h2 * WW + w0 + laneLo];
    acc = __builtin_amdgcn_wmma_f32_16x16x4_f32(false, av[c], false, b, (short)0, acc,
                                                false, false);
  }
#pragma unroll
  for (int v = 0; v < 8; ++v) {
    const int ho = h0 + v + (laneHi << 3);
    out[((size_t)dp * HH + ho) * WW + w0 + laneLo] = acc[v];   // coalesced
  }
}

// ---------------- pass 3 (fused): blur along D + SSIM map + reduce ----------------
// buf2 holds 5 fields of shape (DP, HH, WW); only the interior 374x118x54 is kept.
__global__ __launch_bounds__(256) void ssim_reduce(const float* __restrict__ b2,
                                                   G11 gg, float* __restrict__ ssum) {
  const int DI = DD - 2*PADR, HI = HH - 2*PADR, WI = WW - 2*PADR; // 374,118,54
  const int total = DI * HI * WI;
  const int idx = blockIdx.x * 256 + threadIdx.x;
  float val = 0.0f;
  if (idx < total) {
    const int wi = idx % WI; const int r = idx / WI;
    const int hi = r % HI;   const int di = r / HI;
    const int d = di + PADR, h = hi + PADR, w = wi + PADR;
    const float* p0 = b2 + ((size_t)d * HH + h) * WW + w;
    float mx = 0.f, my = 0.f, exx = 0.f, eyy = 0.f, exy = 0.f;
#pragma unroll
    for (int t = 0; t < 11; ++t) {
      const size_t off = (size_t)t * HH * WW;   // step along padded D
      const float g = gg.g[t];
      mx  += g * p0[off];
      my  += g * p0[off +     B2_ELEMS];
      exx += g * p0[off + 2 * B2_ELEMS];
      eyy += g * p0[off + 3 * B2_ELEMS];
      exy += g * p0[off + 4 * B2_ELEMS];
    }
    const float c1 = 4.0e-4f, c2 = 36.0e-4f;   // (0.01*2)^2, (0.03*2)^2
    const float sxx = exx - mx * mx;
    const float syy = eyy - my * my;
    const float sxy = exy - mx * my;
    val = ((2.f * mx * my + c1) * (2.f * sxy + c2)) /
          ((mx * mx + my * my + c1) * (sxx + syy + c2));
  }
  __shared__ float sh[256];
  sh[threadIdx.x] = val;
  __syncthreads();
  for (int s = 128; s > 0; s >>= 1) {
    if (threadIdx.x < s) sh[threadIdx.x] += sh[threadIdx.x + s];
    __syncthreads();
  }
  if (threadIdx.x == 0) atomicAdd(ssum, sh[0]);
}

// ---------------- axis-normalized sums: one block per bin, no atomics ----------------
__global__ __launch_bounds__(256) void axis_sums(const float* __restrict__ vx, const float* __restrict__ vy,
                                                 const float* __restrict__ vz, const float* __restrict__ pp,
                                                 const float* __restrict__ mk,
                                                 const float* __restrict__ rvx, const float* __restrict__ rvy,
                                                 const float* __restrict__ rvz, const float* __restrict__ rp,
                                                 float* __restrict__ red) {
  const int bin = blockIdx.x;
  const int tid = threadIdx.x;
  float sdf = 0.f, sm = 0.f;
  if (bin < DD) {
    const int d = bin;
    for (int i = tid; i < HH * WW; i += 256) {
      const size_t idx = (size_t)d * HH * WW + i;
      const float a0 = vx[idx]-rvx[idx], a1 = vy[idx]-rvy[idx];
      const float a2 = vz[idx]-rvz[idx], a3 = pp[idx]-rp[idx];
      sdf += a0*a0 + a1*a1 + a2*a2 + a3*a3;
      const float m = mk[idx]; sm += m*m;
    }
  } else if (bin < DD + HH) {
    const int h = bin - DD;
    for (int i = tid; i < DD * WW; i += 256) {
      const int d = i / WW, w = i - (i / WW) * WW;
      const size_t idx = ((size_t)d * HH + h) * WW + w;
      const float a0 = vx[idx]-rvx[idx], a1 = vy[idx]-rvy[idx];
      const float a2 = vz[idx]-rvz[idx], a3 = pp[idx]-rp[idx];
      sdf += a0*a0 + a1*a1 + a2*a2 + a3*a3;
      const float m = mk[idx]; sm += m*m;
    }
  } else {
    const int w = bin - DD - HH;
    for (int i = tid; i < DD * HH; i += 256) {
      const size_t idx = (size_t)i * WW + w;
      const float a0 = vx[idx]-rvx[idx], a1 = vy[idx]-rvy[idx];
      const float a2 = vz[idx]-rvz[idx], a3 = pp[idx]-rp[idx];
      sdf += a0*a0 + a1*a1 + a2*a2 + a3*a3;
      const float m = mk[idx]; sm += m*m;
    }
  }
  __shared__ float s1[256], s2[256];
  s1[tid] = sdf; s2[tid] = sm;
  __syncthreads();
  for (int s = 128; s > 0; s >>= 1) {
    if (tid < s) { s1[tid] += s1[tid + s]; s2[tid] += s2[tid + s]; }
    __syncthreads();
  }
  if (tid == 0) {
    if (bin < DD)            { red[R_DIFFD + bin] = s1[0];            red[R_M2D + bin] = s2[0]; }
    else if (bin < DD + HH)  { red[R_DIFFH + bin - DD] = s1[0];       red[R_M2H + bin - DD] = s2[0]; }
    else                     { red[R_DIFFW + bin - DD - HH] = s1[0];  red[R_M2W + bin - DD - HH] = s2[0]; }
  }
}

// ---------------- dice + MSE streaming reduction ----------------
__global__ __launch_bounds__(256) void dice_mse(const float* __restrict__ t0, const float* __restrict__ t1,
                                                const float* __restrict__ t2, const float* __restrict__ t3,
                                                const float* __restrict__ p0, const float* __restrict__ p1,
                                                const float* __restrict__ p2, const float* __restrict__ p3,
                                                float* __restrict__ scal) {
  const size_t N = 4 * VOL;
  float inter = 0.f, sp = 0.f, st = 0.f, sq = 0.f;
  for (size_t i = (size_t)blockIdx.x * 256 + threadIdx.x; i < N; i += (size_t)gridDim.x * 256) {
    const int c = (int)(i / VOL);
    const size_t off = i - (size_t)c * VOL;
    const float* T = (c == 0) ? t0 : (c == 1) ? t1 : (c == 2) ? t2 : t3;
    const float* P = (c == 0) ? p0 : (c == 1) ? p1 : (c == 2) ? p2 : p3;
    const float tv = T[off], pv = P[off];
    inter += tv * pv; sp += pv * pv; st += tv * tv;
    const float df = pv - tv; sq += df * df;
  }
  __shared__ float sh[256];
  float vals[4] = {inter, sp, st, sq};
  for (int k = 0; k < 4; ++k) {
    sh[threadIdx.x] = vals[k];
    __syncthreads();
    for (int s = 128; s > 0; s >>= 1) {
      if (threadIdx.x < s) sh[threadIdx.x] += sh[threadIdx.x + s];
      __syncthreads();
    }
    if (threadIdx.x == 0) atomicAdd(&scal[k], sh[0]);
    __syncthreads();
  }
}

__global__ void zero_scal(float* s) { if (threadIdx.x < 8) s[threadIdx.x] = 0.f; }

__global__ void finalize(const float* __restrict__ red, float* __restrict__ out) {
  if (threadIdx.x == 0 && blockIdx.x == 0) {
    double sl = 0.0, sw = 0.0, shv = 0.0;
    for (int d = 0; d < DD; ++d) sl  += (double)red[R_DIFFD + d] / (double)red[R_M2D + d];
    for (int h = 0; h < HH; ++h) sw  += (double)red[R_DIFFH + h] / (double)red[R_M2H + h];
    for (int w = 0; w < WW; ++w) shv += (double)red[R_DIFFW + w] / (double)red[R_M2W + w];
    sl /= DD; sw /= HH; shv /= WW;
    const double inter = red[R_SCAL + 0], sp = red[R_SCAL + 1];
    const double st = red[R_SCAL + 2],    sq = red[R_SCAL + 3];
    const double ss = red[R_SCAL + 4];
    const double dice = 1.0 - (2.0 * inter + 1.0) / (sp + st + 1.0);
    const double mse = sq / (4.0 * (double)VOL);
    const double nssim = 4.0 * (double)(DD - 10) * (double)(HH - 10) * (double)(WW - 10);
    const double ssim_loss = 1.0 - ss / nssim;
    out[0] = (float)(dice + ssim_loss + mse + sl + sw + shv);
  }
}

extern "C" void kernel_launch(void* const* d_in, const int* in_sizes, int n_in,
                              void* d_out, int out_size, void* d_ws, size_t ws_size,
                              hipStream_t stream) {
  (void)in_sizes; (void)n_in; (void)out_size; (void)ws_size;
  const float* vx  = (const float*)d_in[0];
  const float* vy  = (const float*)d_in[1];
  const float* vz  = (const float*)d_in[2];
  const float* pp  = (const float*)d_in[3];
  const float* mk  = (const float*)d_in[4];
  const float* rvx = (const float*)d_in[5];
  const float* rvy = (const float*)d_in[6];
  const float* rvz = (const float*)d_in[7];
  const float* rp  = (const float*)d_in[8];

  float* ws    = (float*)d_ws;
  float* buf1a = ws;                      // (DP, HP, WW)
  float* buf1b = ws + B1_ELEMS;           // (DP, HP, WW)
  float* buf2  = ws + 2 * B1_ELEMS;       // 5 x (DP, HH, WW)
  float* red   = ws + RED_OFF;
  float* scal  = red + R_SCAL;
  float* gtab  = red + R_GTAB;

  // Gaussian taps (host-side, deterministic)
  float g[11], norm = 0.f;
  for (int i = 0; i < 11; ++i) { float t = (float)(i - 5); g[i] = expf(-t * t / 4.5f); norm += g[i]; }
  const float invn = 1.0f / norm;
  G11 gg; for (int i = 0; i < 11; ++i) gg.g[i] = g[i] * invn;

  zero_scal<<<1, 32, 0, stream>>>(scal);
  fill_gtab<<<(GTAB_N + 255) / 256, 256, 0, stream>>>(gtab, invn);
  axis_sums<<<DD + HH + WW, 256, 0, stream>>>(vx, vy, vz, pp, mk, rvx, rvy, rvz, rp, red);
  dice_mse<<<2048, 256, 0, stream>>>(vx, vy, vz, pp, rvx, rvy, rvz, rp, scal);

  const float* T[4] = {vx, vy, vz, pp};
  const float* P[4] = {rvx, rvy, rvz, rp};
  const dim3 g1(4, LTILES), g2(32, DP);
  const long interior = (long)(DD - 10) * (HH - 10) * (WW - 10);   // 2,383,128
  for (int ch = 0; ch < 4; ++ch) {
    // fields: 0=mu_x, 1=mu_y, 2=E[x^2], 3=E[y^2], 4=E[xy]  (x=pred, y=target)
    blur_w_wmma2<0, 2><<<g1, 32, 0, stream>>>(P[ch], T[ch], buf1a, buf1b, gtab);
    blur_h_wmma<<<g2, 32, 0, stream>>>(buf1a, buf2 + 0 * B2_ELEMS, gtab);
    blur_h_wmma<<<g2, 32, 0, stream>>>(buf1b, buf2 + 2 * B2_ELEMS, gtab);
    blur_w_wmma2<1, 3><<<g1, 32, 0, stream>>>(P[ch], T[ch], buf1a, buf1b, gtab);
    blur_h_wmma<<<g2, 32, 0, stream>>>(buf1a, buf2 + 1 * B2_ELEMS, gtab);
    blur_h_wmma<<<g2, 32, 0, stream>>>(buf1b, buf2 + 3 * B2_ELEMS, gtab);
    blur_w_wmma1<4><<<g1, 32, 0, stream>>>(P[ch], T[ch], buf1a, gtab);
    blur_h_wmma<<<g2, 32, 0, stream>>>(buf1a, buf2 + 4 * B2_ELEMS, gtab);
    ssim_reduce<<<(unsigned)((interior + 255) / 256), 256, 0, stream>>>(buf2, gg, scal + 4);
  }
  finalize<<<1, 1, 0, stream>>>(red, (float*)d_out);
}